// SmoothnessLoss_90125593739302
// MI455X (gfx1250) — compile-verified
//
#include <hip/hip_runtime.h>
#include <stdint.h>
#include <stddef.h>

// ---------------------------------------------------------------------------
// SmoothnessLoss on MI455X (gfx1250).
// Compute-bound (f32 VALU + transcendentals); inputs fully L2-resident, so
// the optimization targets are LDS reuse (TDM tensor_load_to_lds staging of
// 50x50x3 tiles), native v_exp_f32/v_log_f32, wave32 block shaping, and
// f32 WMMA (16x16x4) for the deterministic reductions.
// ---------------------------------------------------------------------------

#define WSZ   21
#define KW    43               // window length
#define NOFF  (KW * KW)        // 1849 offsets
#define HH    128
#define WW    128
#define BB    2
#define CC    3
#define PH    (HH + 2 * WSZ)   // 170 (reflect-padded)
#define PW    (WW + 2 * WSZ)   // 170
#define TILE  8
#define REG   (TILE + 2 * WSZ) // 50 staged region per tile
#define SIG_COLOR 50.0f
#define SIG_SPACE (1.0f / 98.0f)
#define LOG2E 1.44269504088896340736f

// workspace layout (float elements)
#define WS_PAD_S 0
#define WS_PAD_O (BB * CC * PH * PW)        // 173400
#define WS_PART  (2 * BB * CC * PH * PW)    // 346800
#define NBLK ((WW / TILE) * (HH / TILE) * BB) // 512 partial sums

// ---- CDNA5 feature availability (device pass only; fallbacks keep host ok)
#if defined(__has_builtin)
#if __has_builtin(__builtin_amdgcn_tensor_load_to_lds) && \
    __has_builtin(__builtin_amdgcn_s_wait_tensorcnt)
#define HAVE_TDM 1
#endif
#if __has_builtin(__builtin_amdgcn_wmma_f32_16x16x4_f32)
#define HAVE_WMMA 1
#endif
#endif
#ifndef HAVE_TDM
#define HAVE_TDM 0
#endif
#ifndef HAVE_WMMA
#define HAVE_WMMA 0
#endif

struct Smem {
    float s[CC][REG][REG];   // smooth tile  (30000 B), LDS offset 0
    float o[CC][REG][REG];   // original tile (30000 B), LDS offset 30000
    float tbl[KW + 5];       // exp(-sigma_space * k^2), k=-21..21
    float red[256];
};                            // 61 KB static LDS (<< 320 KB/WGP)

#if HAVE_TDM
typedef unsigned int u32x4 __attribute__((ext_vector_type(4)));
typedef int i32x8 __attribute__((ext_vector_type(8)));
typedef int i32x4 __attribute__((ext_vector_type(4)));

// Issue one TDM 3-D tile load: 50x50x3 tile from a 170x170x3 f32 tensor.
// D# packed per CDNA5 ISA 8.3-8.6 (group0: count/lds/global addr; group1:
// data_size=4B, dims/strides; group2: tensor_dim2=3; group3: zero).
// This toolchain's builtin takes 6 args: the 5th is an extra int32x8 group
// (VADDR4 slot — unused per ISA, "set to NULL"), passed as zeros.
__device__ __forceinline__ void tdm_load_tile(const float* gsrc, unsigned lds_byte_off) {
    uint64_t ga = (uint64_t)(uintptr_t)gsrc;
    u32x4 g0;
    g0[0] = 1u;                                            // count=1 (valid), user mode
    g0[1] = lds_byte_off;                                  // lds_addr [63:32]
    g0[2] = (unsigned)ga;                                  // global_addr lo
    g0[3] = (unsigned)((ga >> 32) & 0x1FFFFFFu) | (2u << 30); // addr[56:32] | type=2
    i32x8 g1;
    g1[0] = (2 << 16);                                     // data_size = 4 bytes
    g1[1] = (PW & 0xFFFF) << 16;                           // tensor_dim0 lo16
    g1[2] = ((PH & 0xFFFF) << 16);                         // dim0 hi16=0 | tensor_dim1 lo16
    g1[3] = (REG << 16);                                   // dim1 hi16=0 | tile_dim0
    g1[4] = REG | (CC << 16);                              // tile_dim1 | tile_dim2
    g1[5] = PW;                                            // tensor_dim0_stride lo32
    g1[6] = ((PH * PW) & 0xFFFF) << 16;                    // stride0 hi16=0 | stride1 lo16
    g1[7] = (PH * PW) >> 16;                               // stride1 hi32
    i32x4 g2; g2[0] = CC; g2[1] = 0; g2[2] = 0; g2[3] = 0; // tensor_dim2=3
    i32x4 g3; g3[0] = 0; g3[1] = 0; g3[2] = 0; g3[3] = 0;
    i32x8 g4 = {0, 0, 0, 0, 0, 0, 0, 0};                   // unused group (VADDR4)
    __builtin_amdgcn_tensor_load_to_lds(g0, g1, g2, g3, g4, 0);
}
#endif

#if HAVE_WMMA
typedef float v2f __attribute__((ext_vector_type(2)));
typedef float v8f __attribute__((ext_vector_type(8)));

// Chained ones(16x4) x B(4x16) + C : D[m][n] = C[m][n] + (column sums of B).
// With all-ones A, the total of any 64 values packed bijectively into B's
// lane/VGPR slots equals the sum of row 0 of D, independent of B's exact
// operand layout. Full f32 precision (V_WMMA_F32_16X16X4_F32), wave32,
// EXEC all-ones (entire wave executes).
__device__ __forceinline__ v8f wmma_accum64(v8f c, float b0, float b1) {
    v2f a; a[0] = 1.0f; a[1] = 1.0f;
    v2f b; b[0] = b0;   b[1] = b1;
    return __builtin_amdgcn_wmma_f32_16x16x4_f32(
        /*neg_a=*/false, a, /*neg_b=*/false, b,
        /*c_mod=*/(short)0, c, /*reuse_a=*/false, /*reuse_b=*/false);
}
#endif

// ---------------------------------------------------------------------------
// Kernel 1: reflect-pad both images into workspace (TDM cannot reflect; OOB
// returns zero, so we materialize the 170x170 padded tensors once — 1.4 MB).
// ---------------------------------------------------------------------------
__global__ void pad_reflect_kernel(const float* __restrict__ orig,
                                   const float* __restrict__ smth,
                                   float* __restrict__ ws) {
    int i = blockIdx.x * blockDim.x + threadIdx.x;
    const int per_img = BB * CC * PH * PW;
    if (i >= 2 * per_img) return;
    int img = i / per_img;
    int r   = i - img * per_img;
    int bc  = r / (PH * PW);
    int p   = r - bc * (PH * PW);
    int pr  = p / PW, pc = p - pr * PW;
    int sr = pr - WSZ; sr = sr < 0 ? -sr : (sr >= HH ? 2 * (HH - 1) - sr : sr);
    int sc = pc - WSZ; sc = sc < 0 ? -sc : (sc >= WW ? 2 * (WW - 1) - sc : sc);
    const float* src = (img == 0) ? smth : orig;
    float* dst = ws + (img == 0 ? WS_PAD_S : WS_PAD_O);
    dst[r] = src[(bc * HH + sr) * WW + sc];
}

// ---------------------------------------------------------------------------
// Kernel 2: main loss. 256 threads = 8 wave32 = 64 pixels x 4 offset-slices.
// ---------------------------------------------------------------------------
__global__ __launch_bounds__(256) void smoothness_main_kernel(
        const float* __restrict__ ws, float* __restrict__ partials) {
    __shared__ Smem sm;
    const int tid = threadIdx.x;
    const int x0 = blockIdx.x * TILE;   // tile origin (image coords == padded-region origin)
    const int y0 = blockIdx.y * TILE;
    const int b  = blockIdx.z;

    const float* pad_s = ws + WS_PAD_S + (size_t)b * CC * PH * PW;
    const float* pad_o = ws + WS_PAD_O + (size_t)b * CC * PH * PW;

#if HAVE_TDM
    // Wave 0 issues two async tensor tile loads, waits TENSORcnt, then the
    // workgroup barrier makes the LDS tiles visible to all 8 waves.
    if (tid < 32) {
        tdm_load_tile(pad_s + (size_t)y0 * PW + x0, (unsigned)offsetof(Smem, s));
        tdm_load_tile(pad_o + (size_t)y0 * PW + x0, (unsigned)offsetof(Smem, o));
        __builtin_amdgcn_s_wait_tensorcnt(0);
    }
#else
    for (int i = tid; i < CC * REG * REG; i += 256) {
        int c  = i / (REG * REG);
        int rr = (i / REG) % REG;
        int cc = i % REG;
        size_t g = ((size_t)c * PH + (y0 + rr)) * PW + (x0 + cc);
        (&sm.s[0][0][0])[i] = pad_s[g];
        (&sm.o[0][0][0])[i] = pad_o[g];
    }
#endif
    if (tid < KW) {
        float k = (float)(tid - WSZ);
        sm.tbl[tid] = __builtin_amdgcn_exp2f(-SIG_SPACE * LOG2E * k * k);
    }
    __syncthreads();

    const int pid = tid & 63;        // pixel within 8x8 tile
    const int z   = tid >> 6;        // offset slice 0..3
    const int ly = pid >> 3, lx = pid & 7;
    const int rr0 = ly + WSZ, cc0 = lx + WSZ;

    float scv[CC], ocv[CC];
#pragma unroll
    for (int c = 0; c < CC; ++c) {
        scv[c] = sm.s[c][rr0][cc0];
        ocv[c] = sm.o[c][rr0][cc0];
    }

    // Sobel edge responses from the staged tiles (reflect already baked in).
    float eo = 0.f, es = 0.f;
#pragma unroll
    for (int c = 0; c < CC; ++c) {
        {   // smooth
            float a = sm.s[c][rr0-1][cc0-1], t = sm.s[c][rr0-1][cc0], u = sm.s[c][rr0-1][cc0+1];
            float l = sm.s[c][rr0  ][cc0-1],                         r = sm.s[c][rr0  ][cc0+1];
            float f = sm.s[c][rr0+1][cc0-1], g = sm.s[c][rr0+1][cc0], h = sm.s[c][rr0+1][cc0+1];
            float gx = (u - a) + 2.f * (r - l) + (h - f);
            float gy = (f - a) + 2.f * (g - t) + (h - u);
            es += __builtin_amdgcn_sqrtf(gx * gx + gy * gy);
        }
        {   // original
            float a = sm.o[c][rr0-1][cc0-1], t = sm.o[c][rr0-1][cc0], u = sm.o[c][rr0-1][cc0+1];
            float l = sm.o[c][rr0  ][cc0-1],                         r = sm.o[c][rr0  ][cc0+1];
            float f = sm.o[c][rr0+1][cc0-1], g = sm.o[c][rr0+1][cc0], h = sm.o[c][rr0+1][cc0+1];
            float gx = (u - a) + 2.f * (r - l) + (h - f);
            float gy = (f - a) + 2.f * (g - t) + (h - u);
            eo += __builtin_amdgcn_sqrtf(gx * gx + gy * gy);
        }
    }
    const float mL = (eo < 20.0f && (es - eo) > 10.0f) ? 1.0f : 0.0f;
    const float mS = 1.0f - mL;

    // 1849-offset loop, split 4 ways across the z slices. Within a wave all
    // lanes share k, so tbl reads are broadcasts and rr/cc stay conflict-free
    // (row stride 50 dwords over 64 banks).
    float acc = 0.0f;
    for (int k = z; k < NOFF; k += 4) {
        int ix = k / KW;                 // row-offset index 0..42
        int iy = k - ix * KW;            // col-offset index 0..42
        float wsw = sm.tbl[ix] * sm.tbl[iy];
        int rr = ly + ix, cc = lx + iy;  // = (ly+WSZ)+(ix-WSZ), in [0,49]
        float sum_d2 = 0.f, sum_dp = 0.f, sum_o2 = 0.f;
#pragma unroll
        for (int c = 0; c < CC; ++c) {
            float dsv = fabsf(scv[c] - sm.s[c][rr][cc]);
            sum_d2 = fmaf(dsv, dsv, sum_d2);
            // safe_pow(d, 0.8): exp2(0.8*log2(d)); log2(0)=-inf -> exp2 = 0
            sum_dp += __builtin_amdgcn_exp2f(0.8f * __builtin_amdgcn_logf(dsv));
            float dov = ocv[c] - sm.o[c][rr][cc];
            sum_o2 = fmaf(dov, dov, sum_o2);
        }
        float wr = __builtin_amdgcn_exp2f(-SIG_COLOR * LOG2E * sum_o2);
        acc += mL * (wsw * sum_d2) + mS * (wr * sum_dp);
    }

    // deterministic block reduction: 256 partials -> scalar
    sm.red[tid] = acc;
    __syncthreads();
#if HAVE_WMMA
    if (tid < 32) {                      // wave 0, EXEC all ones
        v8f c = {};
#pragma unroll
        for (int ch = 0; ch < 4; ++ch)
            c = wmma_accum64(c, sm.red[ch * 64 + tid], sm.red[ch * 64 + 32 + tid]);
        sm.red[tid] = c[0];              // lanes 0..15: D[0][n] = column sums
    }
    __syncthreads();
    if (tid == 0) {
        float t = 0.0f;
        for (int i = 0; i < 16; ++i) t += sm.red[i];
        int blin = (blockIdx.z * gridDim.y + blockIdx.y) * gridDim.x + blockIdx.x;
        partials[blin] = t;
    }
#else
    for (int s = 128; s > 0; s >>= 1) {
        if (tid < s) sm.red[tid] += sm.red[tid + s];
        __syncthreads();
    }
    if (tid == 0) {
        int blin = (blockIdx.z * gridDim.y + blockIdx.y) * gridDim.x + blockIdx.x;
        partials[blin] = sm.red[0];
    }
#endif
}

// ---------------------------------------------------------------------------
// Kernel 3: deterministic finalize (512 partials -> scalar mean), also via
// chained f32 WMMA column-sum reduction.
// ---------------------------------------------------------------------------
__global__ void finalize_kernel(const float* __restrict__ partials,
                                float* __restrict__ out) {
    __shared__ float red[256];
    int tid = threadIdx.x;
#if HAVE_WMMA
    if (tid < 32) {                      // wave 0, EXEC all ones
        v8f c = {};
#pragma unroll
        for (int ch = 0; ch < 8; ++ch)
            c = wmma_accum64(c, partials[ch * 64 + tid], partials[ch * 64 + 32 + tid]);
        red[tid] = c[0];
    }
    __syncthreads();
    if (tid == 0) {
        float t = 0.0f;
        for (int i = 0; i < 16; ++i) t += red[i];
        out[0] = t * (1.0f / (float)(HH * WW * BB));
    }
#else
    red[tid] = partials[tid] + partials[tid + 256];
    __syncthreads();
    for (int s = 128; s > 0; s >>= 1) {
        if (tid < s) red[tid] += red[tid + s];
        __syncthreads();
    }
    if (tid == 0) out[0] = red[0] * (1.0f / (float)(HH * WW * BB));
#endif
}

extern "C" void kernel_launch(void* const* d_in, const int* in_sizes, int n_in,
                              void* d_out, int out_size, void* d_ws, size_t ws_size,
                              hipStream_t stream) {
    (void)in_sizes; (void)n_in; (void)out_size; (void)ws_size;
    const float* orig = (const float*)d_in[0];   // original_images
    const float* smth = (const float*)d_in[1];   // smooth_images
    float* ws  = (float*)d_ws;                   // needs ~1.39 MB
    float* out = (float*)d_out;

    const int npad = 2 * BB * CC * PH * PW;
    pad_reflect_kernel<<<(npad + 255) / 256, 256, 0, stream>>>(orig, smth, ws);

    dim3 grid(WW / TILE, HH / TILE, BB);         // 16 x 16 x 2 = 512 blocks
    smoothness_main_kernel<<<grid, 256, 0, stream>>>(ws, ws + WS_PART);

    finalize_kernel<<<1, 256, 0, stream>>>(ws + WS_PART, out);
}